// RoIPool_64819646432058
// MI455X (gfx1250) — compile-verified
//
#include <hip/hip_runtime.h>
#include <cstdint>
#include <cstddef>

// Problem constants (from the reference): B=8, C=256, H=W=64, N=128 cells.
#define NB    8
#define NC    256
#define NHW   4096      // H*W
#define CPB   64        // channels per block in the max kernel
#define CPT   8         // channels per thread (CPB / (256/32))

// ---- CDNA5 async global->LDS probe -----------------------------------------
#if defined(__gfx1250__) && __has_builtin(__builtin_amdgcn_global_load_async_to_lds_b128)
#define USE_ASYNC_LDS 1
#else
#define USE_ASYNC_LDS 0
#endif

#if USE_ASYNC_LDS
typedef int v4i __attribute__((ext_vector_type(4)));
typedef __attribute__((address_space(1))) v4i global_v4i;
typedef __attribute__((address_space(3))) v4i lds_v4i;
#endif

// ---------------------------------------------------------------------------
// Pass 1: pack per-cell boolean masks (1 byte each) into one uint32 bitfield
// per (image, pixel): bit k = mask of k-th cell of that image at that pixel.
// Reads 512 KB once, writes 128 KB. Removes 256x mask re-decode in pass 2.
// ---------------------------------------------------------------------------
__global__ __launch_bounds__(256) void pack_masks_kernel(
    const unsigned char* __restrict__ masks,   // (N, H*W) bool as u8
    const int* __restrict__ counts,            // (B,) cells per image
    unsigned int* __restrict__ packed)         // (B, H*W) bitfields
{
  int idx = blockIdx.x * blockDim.x + threadIdx.x;
  if (idx >= NB * NHW) return;
  int b  = idx >> 12;           // / NHW
  int hw = idx & (NHW - 1);

  int start = 0;
  #pragma unroll
  for (int i = 0; i < NB; ++i) start += (i < b) ? counts[i] : 0;
  int cnt = counts[b];
  if (cnt > 32) cnt = 32;       // bitfield capacity (problem uses 16)

  unsigned int bits = 0;
  for (int k = 0; k < cnt; ++k)
    bits |= (masks[(size_t)(start + k) * NHW + hw] ? 1u : 0u) << k;
  packed[idx] = bits;
}

// ---------------------------------------------------------------------------
// Pass 2: one block per (cell, 64-channel slice). The block stages its image's
// packed-mask slice (16 KB) into LDS via CDNA5 async global->LDS DMA, then
// each 32-lane wave-slice owns 8 channels: lanes stripe pixels, test one bit,
// and on hits do 8 coalesced loads + 8 v_max_f32. Wave32 shfl-xor max tree.
// ---------------------------------------------------------------------------
__global__ __launch_bounds__(256) void cell_max_kernel(
    const float* __restrict__ fm,              // (B, C, H*W)
    const unsigned int* __restrict__ packed,   // (B, H*W)
    const int* __restrict__ counts,            // (B,)
    float* __restrict__ out)                   // (N, C)
{
  __shared__ unsigned int smask[NHW];          // 16 KB of 320 KB/WGP

  const int n     = blockIdx.x;                // cell id
  const int cbase = blockIdx.y * CPB;          // first channel of this block

  // cell -> (image b, local cell index k), per reference repeat() semantics
  int b = 0, start = 0;
  for (int i = 0; i < NB; ++i) {
    int c = counts[i];
    if (n < start + c) { b = i; break; }
    start += c;
  }
  const int k = n - start;

  const unsigned int* src = packed + (size_t)b * NHW;

#if USE_ASYNC_LDS
  // Async DMA: 256 threads x 4 x b128 = 16 KB, tracked by ASYNCcnt.
  for (int i = threadIdx.x * 4; i < NHW; i += 256 * 4) {
    __builtin_amdgcn_global_load_async_to_lds_b128(
        (global_v4i*)(src + i),
        (lds_v4i*)(&smask[i]),
        0, 0);
  }
#if __has_builtin(__builtin_amdgcn_s_wait_asynccnt)
  __builtin_amdgcn_s_wait_asynccnt(0);
#else
  asm volatile("s_wait_asynccnt 0" ::: "memory");
#endif
#else
  for (int i = threadIdx.x; i < NHW; i += 256) smask[i] = src[i];
#endif
  __syncthreads();

  const int lane = threadIdx.x & 31;
  const int g    = threadIdx.x >> 5;           // channel-group 0..7
  const float* __restrict__ f0 =
      fm + (size_t)(b * NC + cbase + g * CPT) * NHW;

  float acc[CPT];
  #pragma unroll
  for (int j = 0; j < CPT; ++j) acc[j] = -__builtin_inff();

  // Lanes stripe pixels; ~25% of iterations take the hit path.
  for (int hw = lane; hw < NHW; hw += 32) {
    unsigned int m = smask[hw];
    if ((m >> k) & 1u) {
      #pragma unroll
      for (int j = 0; j < CPT; ++j)
        acc[j] = fmaxf(acc[j], f0[(size_t)j * NHW + hw]);
    }
  }

  // wave32 cross-lane max reduction
  #pragma unroll
  for (int off = 16; off > 0; off >>= 1) {
    #pragma unroll
    for (int j = 0; j < CPT; ++j)
      acc[j] = fmaxf(acc[j], __shfl_xor(acc[j], off, 32));
  }

  if (lane == 0) {
    #pragma unroll
    for (int j = 0; j < CPT; ++j)
      out[(size_t)n * NC + cbase + g * CPT + j] = acc[j];
  }
}

// ---------------------------------------------------------------------------
extern "C" void kernel_launch(void* const* d_in, const int* in_sizes, int n_in,
                              void* d_out, int out_size, void* d_ws, size_t ws_size,
                              hipStream_t stream) {
  const float*         fm     = (const float*)d_in[0];         // (8,256,64,64) f32
  const unsigned char* masks  = (const unsigned char*)d_in[1]; // (128,64,64) bool
  const int*           counts = (const int*)d_in[2];           // (8,) i32
  float*               out    = (float*)d_out;                 // (128,256) f32
  unsigned int*        packed = (unsigned int*)d_ws;           // 128 KB scratch

  int ncells = in_sizes[1] / NHW;  // 128

  int total = NB * NHW;
  pack_masks_kernel<<<(total + 255) / 256, 256, 0, stream>>>(masks, counts, packed);

  dim3 grid(ncells, NC / CPB);     // (128, 4)
  cell_max_kernel<<<grid, 256, 0, stream>>>(fm, packed, counts, out);
}